// PlanMapBoundLoss_43379169690015
// MI455X (gfx1250) — compile-verified
//
#include <hip/hip_runtime.h>
#include <math.h>

// ---------------------------------------------------------------------------
// PlanMapBoundLoss on gfx1250 (MI455X).
// Distances d^2(t,n) = |p_t|^2 + (x_n^2+y_n^2) - 2*p_t.(x_n,y_n) computed as
//   D = A(16x4) * B(4x16) + C  via V_WMMA_F32_16X16X4_F32  (wave32 only).
// A rows  (M=t):   [px_t, py_t, 1, 0]       (rows 12..15 zero-padded)
// B cols  (N=pt):  [-2x, -2y, x^2+y^2, 0]
// C rows:          |p_t|^2
// ---------------------------------------------------------------------------

typedef __attribute__((ext_vector_type(2))) float v2f;
typedef __attribute__((ext_vector_type(8))) float v8f;

#define B_    128
#define T_    12
#define V_    512
#define P_    40
#define GPB   (V_ * P_)     // 20480 points per batch
#define NTILE (GPB / 16)    // 1280 WMMA tiles per batch
#define CHUNKS 4            // blocks per batch in kernel 1
#define WPB    8            // waves per 256-thread block
#define NWAVES (CHUNKS * WPB)

#define MAP_THRESH 0.5f
#define DIS_THRESH 1.0f
#define MASK_FILL  1000000.0f

__global__ void k_init(unsigned long long* __restrict__ packed) {
    int i = blockIdx.x * blockDim.x + threadIdx.x;
    if (i < B_ * T_) packed[i] = 0xFFFFFFFFFFFFFFFFull;
}

__global__ __launch_bounds__(256)
void k_mindist(const float* __restrict__ ego,
               const float* __restrict__ lanes,
               const float* __restrict__ scores,
               unsigned long long* __restrict__ packed)
{
    const int b = blockIdx.y;

    __shared__ float s_px[16], s_py[16], s_sq[16];
    if (threadIdx.x == 0) {
        float cx = 0.f, cy = 0.f;
        for (int t = 0; t < 16; ++t) {
            if (t < T_) { cx += ego[(b * T_ + t) * 2 + 0]; cy += ego[(b * T_ + t) * 2 + 1]; }
            s_px[t] = (t < T_) ? cx : 0.f;
            s_py[t] = (t < T_) ? cy : 0.f;
            s_sq[t] = (t < T_) ? (cx * cx + cy * cy) : 0.f;
        }
    }
    __syncthreads();

    const int  lane = threadIdx.x & 31;
    const int  col  = lane & 15;
    const bool lo   = lane < 16;
    const int  wave = blockIdx.x * WPB + (threadIdx.x >> 5);

    // A operand (16x4 f32): lanes 0-15 hold K0/K1 = (px_M, py_M); lanes 16-31 hold K2/K3 = (1,0)
    v2f A;
    A.x = lo ? s_px[col] : 1.0f;
    A.y = lo ? s_py[col] : 0.0f;

    // C operand: row M=t gets |p_t|^2 ; lanes 0-15 carry rows 0..7, lanes 16-31 rows 8..15
    const int bt = lo ? 0 : 8;
    v8f C;
#pragma unroll
    for (int r = 0; r < 8; ++r) C[r] = s_sq[bt + r];

    float    rmin[8];
    unsigned rv[8];
#pragma unroll
    for (int r = 0; r < 8; ++r) { rmin[r] = 3.0e38f; rv[r] = 0u; }

    const float2* __restrict__ lane_xy = (const float2*)lanes;

    for (int tile = wave; tile < NTILE; tile += NWAVES) {
        const int g = tile * 16 + col;      // global point index, increasing per wave
        const int v = g / P_;               // lane instance index
        float2 lp = lane_xy[b * GPB + g];
        float  x  = __builtin_fmaf(lp.x, 30.f, -15.f);
        float  y  = __builtin_fmaf(lp.y, 60.f, -30.f);
        const bool nb = scores[(b * V_ + v) * 3 + 2] < MAP_THRESH;
        x = nb ? MASK_FILL : x;
        y = nb ? MASK_FILL : y;

        // B operand (4x16 f32): lanes 0-15 hold K0/K1 = (-2x, -2y); lanes 16-31 hold K2/K3 = (x^2+y^2, 0)
        v2f Bv;
        Bv.x = lo ? (-2.f * x) : __builtin_fmaf(x, x, y * y);
        Bv.y = lo ? (-2.f * y) : 0.0f;

        v8f D = __builtin_amdgcn_wmma_f32_16x16x4_f32(
            /*neg_a=*/false, A, /*neg_b=*/false, Bv,
            /*c_mod=*/(short)0, C, /*reuse_a=*/false, /*reuse_b=*/false);

#pragma unroll
        for (int r = 0; r < 8; ++r) {
            const float dv     = D[r];
            const bool  better = dv < rmin[r];
            rv[r]   = better ? (unsigned)v : rv[r];
            rmin[r] = better ? dv : rmin[r];
        }
    }

    // Reduce across the 16 lanes of each half (xor masks 1,2,4,8 stay within a half).
#pragma unroll
    for (int m = 1; m < 16; m <<= 1) {
#pragma unroll
        for (int r = 0; r < 8; ++r) {
            const float    od = __shfl_xor(rmin[r], m, 32);
            const unsigned ov = __shfl_xor(rv[r],   m, 32);
            if (od < rmin[r] || (od == rmin[r] && ov < rv[r])) { rmin[r] = od; rv[r] = ov; }
        }
    }

    // d^2 >= 0 so its float bits are order-preserving; low 32 bits = v gives lowest-v tie-break.
    if (lane == 0) {
#pragma unroll
        for (int r = 0; r < 8; ++r) {
            unsigned long long pk = ((unsigned long long)__float_as_uint(rmin[r]) << 32) | rv[r];
            atomicMin(&packed[b * T_ + r], pk);               // t = 0..7
        }
    }
    if (lane == 16) {
#pragma unroll
        for (int r = 0; r < 4; ++r) {
            unsigned long long pk = ((unsigned long long)__float_as_uint(rmin[r]) << 32) | rv[r];
            atomicMin(&packed[b * T_ + 8 + r], pk);           // t = 8..11 (rows 12..15 unused)
        }
    }
}

__global__ __launch_bounds__(512)
void k_finalize(const float* __restrict__ ego,
                const float* __restrict__ lanes,
                const float* __restrict__ scores,
                const float* __restrict__ weight,
                const unsigned long long* __restrict__ packed,
                float* __restrict__ partial)
{
    const int b   = blockIdx.x;
    const int tid = threadIdx.x;

    __shared__ float s_x[T_ + 1], s_y[T_ + 1];   // s_x[t] = start, s_x[t+1] = end of ego segment t
    __shared__ int   s_cross[T_];
    __shared__ float s_loss[T_];

    if (tid == 0) {
        s_x[0] = 0.f; s_y[0] = 0.f;
        float cx = 0.f, cy = 0.f;
        for (int t = 0; t < T_; ++t) {
            cx += ego[(b * T_ + t) * 2 + 0];
            cy += ego[(b * T_ + t) * 2 + 1];
            s_x[t + 1] = cx; s_y[t + 1] = cy;
        }
    }
    if (tid < T_) {
        s_cross[tid] = 0;
        const unsigned long long pk = packed[b * T_ + tid];
        const float md = sqrtf(__uint_as_float((unsigned)(pk >> 32)));
        s_loss[tid] = (md <= DIS_THRESH) ? (DIS_THRESH - md) : 0.f;
    }
    __syncthreads();

    if (tid < T_ * (P_ - 1)) {
        const int t = tid / (P_ - 1);
        const int s = tid % (P_ - 1);
        const unsigned v = (unsigned)(packed[b * T_ + t] & 0xFFFFFFFFull);
        const bool nb = scores[(b * V_ + v) * 3 + 2] < MAP_THRESH;
        const float2* lp = (const float2*)lanes + ((b * V_ + v) * P_ + s);
        const float2 p0 = lp[0], p1 = lp[1];
        float x1 = nb ? MASK_FILL : __builtin_fmaf(p0.x, 30.f, -15.f);
        float y1 = nb ? MASK_FILL : __builtin_fmaf(p0.y, 60.f, -30.f);
        float x2 = nb ? MASK_FILL : __builtin_fmaf(p1.x, 30.f, -15.f);
        float y2 = nb ? MASK_FILL : __builtin_fmaf(p1.y, 60.f, -30.f);

        const float d1x = s_x[t + 1] - s_x[t], d1y = s_y[t + 1] - s_y[t];   // ego segment
        const float d2x = x2 - x1,             d2y = y2 - y1;               // boundary segment
        const float det = d1x * d2y - d2x * d1y;
        if (det != 0.f) {
            const float dx = x1 - s_x[t], dy = y1 - s_y[t];
            const float t1 = (dx * d2y - dy * d2x) / det;
            const float t2 = (dx * d1y - dy * d1x) / det;
            if (t1 >= 0.f && t1 <= 1.f && t2 >= 0.f && t2 <= 1.f)
                atomicOr(&s_cross[t], 1);
        }
    }
    __syncthreads();

    if (tid == 0) {
        int crossed = 0;
        float sum = 0.f;
        for (int t = 0; t < T_; ++t) {
            crossed |= s_cross[t];
            const float l = crossed ? 0.f : s_loss[t];
            sum += l * weight[b * T_ + t];
        }
        partial[b] = sum;
    }
}

__global__ void k_sum(const float* __restrict__ partial, float* __restrict__ out) {
    if (blockIdx.x == 0 && threadIdx.x == 0) {
        float s = 0.f;
        for (int i = 0; i < B_; ++i) s += partial[i];   // deterministic serial sum (128 values)
        out[0] = s * (1.0f / (float)(B_ * T_));         // LOSS_WEIGHT * mean
    }
}

extern "C" void kernel_launch(void* const* d_in, const int* in_sizes, int n_in,
                              void* d_out, int out_size, void* d_ws, size_t ws_size,
                              hipStream_t stream)
{
    const float* ego    = (const float*)d_in[0];   // (128,12,2)
    const float* lanes  = (const float*)d_in[1];   // (128,512,40,2)
    const float* scores = (const float*)d_in[2];   // (128,512,3)
    const float* weight = (const float*)d_in[3];   // (128,12)

    unsigned long long* packed  = (unsigned long long*)d_ws;   // 1536 x u64
    float*              partial = (float*)(packed + B_ * T_);  // 128 x f32

    k_init<<<(B_ * T_ + 255) / 256, 256, 0, stream>>>(packed);
    k_mindist<<<dim3(CHUNKS, B_), 256, 0, stream>>>(ego, lanes, scores, packed);
    k_finalize<<<B_, 512, 0, stream>>>(ego, lanes, scores, weight, packed, partial);
    k_sum<<<1, 32, 0, stream>>>(partial, (float*)d_out);
}